// ROIPool_81003083202761
// MI455X (gfx1250) — compile-verified
//
#include <hip/hip_runtime.h>
#include <math.h>

#ifndef __has_builtin
#define __has_builtin(x) 0
#endif

#if __has_builtin(__builtin_amdgcn_global_load_async_to_lds_b32)
#define ROI_HAS_ASYNC_LDS 1
#else
#define ROI_HAS_ASYNC_LDS 0
#endif

#define AS1 __attribute__((address_space(1)))
#define AS3 __attribute__((address_space(3)))

namespace {
constexpr int   kC     = 128;
constexpr int   kH     = 64;
constexpr int   kW     = 64;
constexpr int   kOutH  = 7;
constexpr int   kOutW  = 7;
constexpr float kScale = 0.0625f;
constexpr int   kBins  = kOutH * kOutW;   // 49
constexpr int   kPerRoi = kC * kBins;     // 6272
}

// One block-column per ROI (blockIdx.x = m). blockIdx.y splits the 6272
// outputs of a ROI across 4 blocks for occupancy. ROI decode is wave-uniform
// -> SALU/s_load; per-output bin windows are small gathers that hit L2
// (feature map is 4MB, resident).
__global__ __launch_bounds__(256) void roi_pool_kernel(
    const float* __restrict__ feat,
    const float* __restrict__ rois,
    float* __restrict__ out) {
  const int m = blockIdx.x;

  float r0, r1, r2, r3, r4;
#if ROI_HAS_ASYNC_LDS
  // CDNA5 async copy path: DMA the 5 ROI floats into LDS, tracked by
  // ASYNCcnt; one lane per dword. Other waves see the data after the
  // issuing wave drains ASYNCcnt and everyone passes the barrier.
  __shared__ float sroi[5];
  if (threadIdx.x < 5) {
    // Builtin wants (addrspace(1) int*, addrspace(3) int*, imm, imm).
    float* gnc = const_cast<float*>(rois) + (size_t)m * 5 + threadIdx.x;
    AS1 int* gsrc = (AS1 int*)gnc;
    AS3 int* ldst = (AS3 int*)&sroi[threadIdx.x];
    __builtin_amdgcn_global_load_async_to_lds_b32(gsrc, ldst, 0, 0);
  }
  asm volatile("s_wait_asynccnt 0" ::: "memory");
  __syncthreads();
  r0 = sroi[0]; r1 = sroi[1]; r2 = sroi[2]; r3 = sroi[3]; r4 = sroi[4];
#else
  const float* r = rois + (size_t)m * 5;
  r0 = r[0]; r1 = r[1]; r2 = r[2]; r3 = r[3]; r4 = r[4];
#endif

  // Match reference exactly: truncate-toward-zero after f32 multiply by
  // the (exact power-of-two) spatial scale. All values are non-negative.
  const int n  = (int)r0;
  const int x1 = (int)(r1 * kScale);
  const int y1 = (int)(r2 * kScale);
  const int x2 = (int)(r3 * kScale);
  const int y2 = (int)(r4 * kScale);
  const int h  = y2 - y1;
  const int w  = x2 - x1;

  const float* __restrict__ g = feat + (size_t)n * (kC * kH * kW);
  float* __restrict__ outm = out + (size_t)m * kPerRoi;

  const int stride = blockDim.x * gridDim.y;           // 1024
  int idx = threadIdx.x + blockIdx.y * blockDim.x;

  for (; idx < kPerRoi; idx += stride) {
    const int j = idx % kOutW;
    const int t = idx / kOutW;
    const int i = t % kOutH;
    const int c = t / kOutH;

    // Bin window [hs,he) x [ws,we): floor start, ceil end (reference's
    // i*h//7 and ((i+1)*h + 6)//7). All operands non-negative.
    const int hs = y1 + (i * h) / kOutH;
    const int he = y1 + ((i + 1) * h + kOutH - 1) / kOutH;
    const int ws = x1 + (j * w) / kOutW;
    const int we = x1 + ((j + 1) * w + kOutW - 1) / kOutW;

    const float* __restrict__ gc = g + (size_t)c * (kH * kW);

    // Prefetch the next iteration's channel region start (global_prefetch_b8)
    // while this bin reduces.
    const int nidx = idx + stride;
    if (nidx < kPerRoi) {
      const int nc = nidx / kBins;
      __builtin_prefetch(g + (size_t)nc * (kH * kW) + y1 * kW + x1, 0, 3);
    }

    float mx = -INFINITY;
    for (int y = hs; y < he; ++y) {
      const float* __restrict__ row = gc + y * kW;
      for (int x = ws; x < we; ++x) {
        mx = fmaxf(mx, row[x]);
      }
    }
    // Empty bins stay -inf -> 0 (reference zeroes non-finite outputs).
    outm[idx] = isfinite(mx) ? mx : 0.0f;
  }
}

extern "C" void kernel_launch(void* const* d_in, const int* in_sizes, int n_in,
                              void* d_out, int out_size, void* d_ws, size_t ws_size,
                              hipStream_t stream) {
  (void)n_in; (void)out_size; (void)d_ws; (void)ws_size;
  const float* feat = (const float*)d_in[0];
  const float* rois = (const float*)d_in[1];
  float* out = (float*)d_out;

  const int M = in_sizes[1] / 5;   // 256 rois
  dim3 grid((unsigned)M, 4, 1);
  dim3 block(256, 1, 1);
  roi_pool_kernel<<<grid, block, 0, stream>>>(feat, rois, out);
}